// WeightedRaySampler_60404420051287
// MI455X (gfx1250) — compile-verified
//
#include <hip/hip_runtime.h>
#include <hip/hip_bf16.h>
#include <stdint.h>

// Problem geometry (fixed by the reference harness)
#define BB 8
#define RR 16384
#define SS 64          // z_vals per ray
#define NSAMP 128      // fine samples
#define SN 192         // SS + NSAMP
#define MBINS 63       // cdf / bins length
#define EPSW 1e-5f
#define WAVES_PER_BLOCK 8
#define GTOT (BB * RR) // 131072 rays

struct __align__(16) WaveShm {
  float  wts[64];    // weights row (async-staged), use [1..62]
  float  zv[64];     // z_vals row (async-staged)
  float2 cz[64];     // (cdf, midpoint) pairs; cz[63].x = +inf sentinel
  float  zs[128];    // inverse-CDF samples (sorted)
  float  zall[192];  // merged sorted depths
};

// Generic (flat) shared pointer -> raw 32-bit LDS byte offset for inline asm.
__device__ __forceinline__ unsigned lds_off(const void* p) {
  return (unsigned)(uintptr_t)(__attribute__((address_space(3))) const void*)p;
}

// Intra-wave LDS ordering: drain DS ops; no barrier needed (single instr stream).
__device__ __forceinline__ void wave_lds_sync() {
  asm volatile("s_wait_dscnt 0x0" ::: "memory");
}

// Largest idx in [0,62] with cz[idx].x <= u (cz[0].x==0, cz[63].x==+inf).
__device__ __forceinline__ int search_cz(const float2* cz, float u) {
  int idx = 0;
#pragma unroll
  for (int step = 32; step >= 1; step >>= 1) {
    const int p = idx + step;            // max reachable probe is 63 (sentinel)
    idx = (cz[p].x <= u) ? p : idx;
  }
  return idx;
}

// |{ i in [0,127] : zs[i] < x }|  (monotone predicate, clamped probes)
__device__ __forceinline__ int cnt_lt128(const float* zs, float x) {
  int idx = -1;
#pragma unroll
  for (int step = 128; step >= 1; step >>= 1) {
    int p = idx + step; p = (p > 127) ? 127 : p;
    idx = (zs[p] < x) ? p : idx;
  }
  return idx + 1;
}

// |{ j in [0,63] : zv[j] <= x }|
__device__ __forceinline__ int cnt_le64(const float* zv, float x) {
  int idx = -1;
#pragma unroll
  for (int step = 64; step >= 1; step >>= 1) {
    int p = idx + step; p = (p > 63) ? 63 : p;
    idx = (zv[p] <= x) ? p : idx;
  }
  return idx + 1;
}

__global__ __launch_bounds__(256) void wrs_kernel(
    const float* __restrict__ rays_d, const float* __restrict__ rays_o,
    const float* __restrict__ z_vals, const float* __restrict__ weights,
    float* __restrict__ out)
{
  __shared__ WaveShm shm[WAVES_PER_BLOCK];
  const int wave = threadIdx.x >> 5;
  const int lane = threadIdx.x & 31;
  const int g    = blockIdx.x * WAVES_PER_BLOCK + wave; // flat (b,r)
  const int r    = g & (RR - 1);
  WaveShm& w = shm[wave];

  // ---- Stage A: async-stage weights row + z_vals row into LDS -----------
  // Lanes 0..15 move 16B each of weights[g,:]; lanes 16..31 of z_vals[r,:].
  {
    const int  l16 = lane & 15;
    const bool isw = lane < 16;
    const unsigned long long gw =
        (unsigned long long)(uintptr_t)(weights + (size_t)g * SS);
    const unsigned long long gz =
        (unsigned long long)(uintptr_t)(z_vals + (size_t)r * SS);
    const unsigned long long ga =
        (isw ? gw : gz) + (unsigned long long)(l16 * 16);
    const unsigned la =
        (isw ? lds_off(w.wts) : lds_off(w.zv)) + l16 * 16;
    asm volatile("global_load_async_to_lds_b128 %0, %1, off"
                 :: "v"(la), "v"(ga) : "memory");
  }
  // Overlap the async-DMA latency with the ray o/d loads (used in Stage F).
  const float* ro = rays_o + (size_t)g * 3;
  const float* rd = rays_d + (size_t)g * 3;
  const float o0 = ro[0], o1 = ro[1], o2 = ro[2];
  const float d0 = rd[0], d1 = rd[1], d2 = rd[2];
  asm volatile("s_wait_asynccnt 0x0" ::: "memory");

  // ---- Stage B: (cdf, midpoint) pairs via wave32 shuffle scan ------------
  const int   j0  = 2 * lane, j1 = j0 + 1;        // j0 in [0,62], j1 in [1,63]
  const float zv0 = w.zv[j0];
  const float zv1 = w.zv[j0 + 1];
  {
    const float zv2 = w.zv[(j0 + 2 < 64) ? j0 + 2 : 63];   // lane31: unused
    const float zh0 = 0.5f * (zv0 + zv1);
    const float zh1 = 0.5f * (zv1 + zv2);
    // pdf numerators: only indices 1..62 contribute
    const float a = (j0 >= 1) ? (w.wts[j0] + EPSW) : 0.0f;
    const float b = (j1 <= 62) ? (w.wts[j1] + EPSW) : 0.0f;
    const float p = a + b;
    float incl = p;                                // inclusive scan of partials
#pragma unroll
    for (int d = 1; d < 32; d <<= 1) {
      const float y = __shfl_up(incl, d, 32);
      if (lane >= d) incl += y;
    }
    const float prefix = incl - p;                 // exclusive prefix
    const float total  = __shfl(incl, 31, 32);     // full (eps-padded) sum
    const float inv    = 1.0f / total;
    w.cz[j0] = make_float2((prefix + a) * inv, zh0);         // cz[0].x == 0
    w.cz[j1] = (lane == 31) ? make_float2(__builtin_inff(), 0.0f)
                            : make_float2((prefix + p) * inv, zh1);
  }
  wave_lds_sync();

  // ---- Stage D: inverse-CDF sampling, lane owns k = 4L..4L+3 -------------
  // 'below' is monotone in u: one branchless search, then tiny advances.
  const int k0 = lane * 4;
  float smp[4];
  {
    int below = search_cz(w.cz, (float)k0 * (1.0f / 127.0f));
#pragma unroll
    for (int j = 0; j < 4; ++j) {
      const float u = (float)(k0 + j) * (1.0f / 127.0f);
      while (w.cz[below + 1].x <= u) ++below;      // stops at sentinel (<=62)
      const int above = (below < MBINS - 1) ? below + 1 : MBINS - 1;
      const float2 lo = w.cz[below];
      const float2 hi = w.cz[above];
      float den = hi.x - lo.x; if (den < 1e-5f) den = 1.0f;
      const float tt = (u - lo.x) / den;
      smp[j] = lo.y + tt * (hi.y - lo.y);
    }
    *reinterpret_cast<float4*>(&w.zs[k0]) =
        make_float4(smp[0], smp[1], smp[2], smp[3]);         // one b128 store
  }
  wave_lds_sync();

  // ---- Stage E: merge two sorted lists via rank scatter ------------------
  // rank(zv[j]) = j + |{zs <  zv[j]}| ; rank(zs[k]) = k + |{zv <= zs[k]}|
  {
    int c = cnt_lt128(w.zs, zv0);                  // zv side: j0 then advance
    w.zall[j0 + c] = zv0;
    while (c < NSAMP && w.zs[c] < zv1) ++c;        // monotone advance
    w.zall[j1 + c] = zv1;
  }
  {
    int c = cnt_le64(w.zv, smp[0]);                // zs side: registers reused
    w.zall[k0 + c] = smp[0];
#pragma unroll
    for (int j = 1; j < 4; ++j) {
      const float x = smp[j];
      while (c < SS && w.zv[c] <= x) ++c;          // monotone advance
      w.zall[k0 + j + c] = x;
    }
  }
  wave_lds_sync();   // zall committed to LDS RAM before DMA reads it

  // ---- Stage F1: z_all out via async LDS->global (overlaps with F2) -----
  float* zo = out + (long long)GTOT * SN * 3 + (size_t)g * SN;
  {
    const unsigned            zl = lds_off(w.zall);
    const unsigned long long  zg = (unsigned long long)(uintptr_t)zo;
#pragma unroll
    for (int t = 0; t < 3; ++t) {
      const unsigned           off = (unsigned)((t * 64 + lane * 2) * 4);
      const unsigned           la  = zl + off;
      const unsigned long long ga  = zg + off;
      asm volatile("global_store_async_from_lds_b64 %0, %1, off"
                   :: "v"(ga), "v"(la) : "memory");
    }
  }

  // ---- Stage F2: pts with wide coalesced stores (4x b128 + 1x b64) ------
  float* po = out + (size_t)g * (SN * 3);          // 576 floats, 16B-aligned
#pragma unroll
  for (int t = 0; t < 4; ++t) {
    const int e = t * 128 + lane * 4;              // 4 consecutive elements
    float v[4];
#pragma unroll
    for (int j = 0; j < 4; ++j) {
      const int i = e + j;
      const int k = i / 3, c = i - 3 * k;
      const float o = (c == 0) ? o0 : ((c == 1) ? o1 : o2);
      const float d = (c == 0) ? d0 : ((c == 1) ? d1 : d2);
      v[j] = fmaf(d, w.zall[k], o);
    }
    reinterpret_cast<float4*>(po)[e >> 2] = make_float4(v[0], v[1], v[2], v[3]);
  }
  {                                                 // tail: elements 512..575
    const int e = 512 + lane * 2;
    float v[2];
#pragma unroll
    for (int j = 0; j < 2; ++j) {
      const int i = e + j;
      const int k = i / 3, c = i - 3 * k;
      const float o = (c == 0) ? o0 : ((c == 1) ? o1 : o2);
      const float d = (c == 0) ? d0 : ((c == 1) ? d1 : d2);
      v[j] = fmaf(d, w.zall[k], o);
    }
    reinterpret_cast<float2*>(po)[e >> 1] = make_float2(v[0], v[1]);
  }

  // Drain async LDS->global stores (s_endpgm's implicit wait-idle also covers).
  asm volatile("s_wait_asynccnt 0x0" ::: "memory");
}

extern "C" void kernel_launch(void* const* d_in, const int* in_sizes, int n_in,
                              void* d_out, int out_size, void* d_ws, size_t ws_size,
                              hipStream_t stream) {
  const float* rays_d  = (const float*)d_in[0];
  const float* rays_o  = (const float*)d_in[1];
  const float* z_vals  = (const float*)d_in[2];
  const float* weights = (const float*)d_in[3];
  float* out = (float*)d_out;
  dim3 grid(GTOT / WAVES_PER_BLOCK);
  dim3 block(WAVES_PER_BLOCK * 32);
  hipLaunchKernelGGL(wrs_kernel, grid, block, 0, stream,
                     rays_d, rays_o, z_vals, weights, out);
}